// Displacement_3100966388189
// MI455X (gfx1250) — compile-verified
//
#include <hip/hip_runtime.h>

typedef float v2f __attribute__((ext_vector_type(2)));
typedef float v8f __attribute__((ext_vector_type(8)));

#define CUT 32
#define XTOT 32768L          // 32*32*32 trailing dims
#define TILE_X 64
#define LDS_STRIDE 72        // 2*72 = 144 ≡ 16 (mod 64 banks) -> hi/lo half-wave bank groups disjoint

__device__ __forceinline__ float2 cmul(float2 a, float2 b) {
    return make_float2(a.x * b.x - a.y * b.y, a.x * b.y + a.y * b.x);
}

// ---------------------------------------------------------------------------
// Kernel 1: build D[b] (32x32 complex) with one wave per batch, lane = row m.
// Stores real/imag planes with layout [b][n][m] (n = column / K index).
// ---------------------------------------------------------------------------
__global__ __launch_bounds__(32)
void disp_matrix_kernel(const float* __restrict__ r, const float* __restrict__ phi,
                        float* __restrict__ Dre, float* __restrict__ Dim) {
    const int b = blockIdx.x;
    const int m = threadIdx.x;              // 0..31
    const float rb = r[b], pb = phi[b];
    const float cp = cosf(pb), sp = sinf(pb);
    const float2 alpha0 = make_float2( rb * cp, rb * sp);   //  r e^{+i phi}
    const float2 alpha1 = make_float2(-rb * cp, rb * sp);   // -r e^{-i phi}
    const float  sqm    = sqrtf((float)m);

    // Column 0: cumprod of alpha0/sqrt(m) scaled by exp(-r^2/2)
    float2 val;
    if (m == 0) {
        val = make_float2(1.0f, 0.0f);
    } else {
        const float inv = 1.0f / sqm;
        val = make_float2(alpha0.x * inv, alpha0.y * inv);
    }
    #pragma unroll
    for (int off = 1; off < 32; off <<= 1) {
        const float px = __shfl_up(val.x, off, 32);
        const float py = __shfl_up(val.y, off, 32);
        if (m >= off) val = cmul(make_float2(px, py), val);
    }
    const float e = expf(-0.5f * rb * rb);
    float2 cur = make_float2(val.x * e, val.y * e);

    float* dre = Dre + b * CUT * CUT;
    float* dim = Dim + b * CUT * CUT;
    dre[m] = cur.x;
    dim[m] = cur.y;

    // Columns n >= 1: D[m,n] = (alpha1*D[m,n-1] + sqrt(m)*D[m-1,n-1]) / sqrt(n)
    for (int n = 1; n < CUT; ++n) {
        const float px = __shfl_up(cur.x, 1, 32);   // lane0: own value, killed by sqm==0
        const float py = __shfl_up(cur.y, 1, 32);
        const float2 t = cmul(alpha1, cur);
        const float inv = 1.0f / sqrtf((float)n);
        cur.x = (t.x + sqm * px) * inv;
        cur.y = (t.y + sqm * py) * inv;
        dre[n * CUT + m] = cur.x;
        dim[n * CUT + m] = cur.y;
    }
}

// ---------------------------------------------------------------------------
// Kernel 2: out[b,m,x] = sum_n D[b,m,n] * s[b,n,x], complex = two real GEMMs
// sharing the B operand. One 16x16 complex tile per wave via
// v_wmma_f32_16x16x4_f32 chained over K=32 (8 chunks).
// ---------------------------------------------------------------------------
__global__ __launch_bounds__(256)
void disp_apply_kernel(const float* __restrict__ state,
                       const float* __restrict__ Dre, const float* __restrict__ Dim,
                       float* __restrict__ out) {
    __shared__ float sb[CUT * LDS_STRIDE];

    const int  b   = blockIdx.y;
    const long x0  = (long)blockIdx.x * TILE_X;
    const int  tid = threadIdx.x;

    // --- stage 32 x 64 state tile into LDS (coalesced float4 loads) ---
    const float* sp = state + (long)b * CUT * XTOT;     // s[b][n][x]
    #pragma unroll
    for (int i = 0; i < 2; ++i) {
        const int f   = i * 256 + tid;                  // float4 slot 0..511
        const int row = f >> 4;                         // 0..31
        const int c4  = f & 15;                         // 0..15
        const float4 v = *(const float4*)(sp + (long)row * XTOT + x0 + c4 * 4);
        *(float4*)(&sb[row * LDS_STRIDE + c4 * 4]) = v;
    }
    __syncthreads();

    const int w     = tid >> 5;        // wave 0..7
    const int lane  = tid & 31;
    const int hi    = lane >> 4;       // half-wave select (K offset for A/B)
    const int ln    = lane & 15;
    const int mhalf = w >> 2;          // 0..1  -> M tile
    const int xsub  = w & 3;           // 0..3  -> X sub-tile

    // --- preload A operands (16x4 f32 layout: M = lane&15, K = 4c + 2*hi + {0,1}) ---
    const float* dre = Dre + b * CUT * CUT;
    const float* dim = Dim + b * CUT * CUT;
    const int mcol = mhalf * 16 + ln;
    v2f a_re[8], a_im[8];
    #pragma unroll
    for (int c = 0; c < 8; ++c) {
        const int k = c * 4 + 2 * hi;
        a_re[c][0] = dre[(k    ) * CUT + mcol];
        a_re[c][1] = dre[(k + 1) * CUT + mcol];
        a_im[c][0] = dim[(k    ) * CUT + mcol];
        a_im[c][1] = dim[(k + 1) * CUT + mcol];
    }

    // --- K loop: 8 chunks x (re, im) WMMAs, shared B operand from LDS ---
    v8f acc_re = {};
    v8f acc_im = {};
    const int col = xsub * 16 + ln;
    #pragma unroll
    for (int c = 0; c < 8; ++c) {
        const int k = c * 4 + 2 * hi;
        v2f bv;
        bv[0] = sb[(k    ) * LDS_STRIDE + col];
        bv[1] = sb[(k + 1) * LDS_STRIDE + col];
        acc_re = __builtin_amdgcn_wmma_f32_16x16x4_f32(
            false, a_re[c], false, bv, (short)0, acc_re, false, false);
        acc_im = __builtin_amdgcn_wmma_f32_16x16x4_f32(
            false, a_im[c], false, bv, (short)0, acc_im, false, false);
    }

    // --- store: C layout M = r + 8*hi, N = lane&15; interleave {re,im} as complex64 ---
    const long xg = x0 + xsub * 16 + ln;
    #pragma unroll
    for (int rr = 0; rr < 8; ++rr) {
        const int m = mhalf * 16 + rr + 8 * hi;
        float2 v = make_float2(acc_re[rr], acc_im[rr]);
        *(float2*)(out + 2 * (((long)b * CUT + m) * XTOT + xg)) = v;
    }
}

extern "C" void kernel_launch(void* const* d_in, const int* in_sizes, int n_in,
                              void* d_out, int out_size, void* d_ws, size_t ws_size,
                              hipStream_t stream) {
    const float* r     = (const float*)d_in[0];
    const float* phi   = (const float*)d_in[1];
    const float* state = (const float*)d_in[2];
    const int B = in_sizes[0];                 // 32

    float* Dre = (float*)d_ws;                 // [B][32][32]
    float* Dim = Dre + (size_t)B * CUT * CUT;
    float* out = (float*)d_out;                // interleaved complex64

    disp_matrix_kernel<<<B, 32, 0, stream>>>(r, phi, Dre, Dim);

    dim3 grid((unsigned)(XTOT / TILE_X), (unsigned)B);   // 512 x 32
    disp_apply_kernel<<<grid, 256, 0, stream>>>(state, Dre, Dim, out);
}